// SphericalFeaturePropagation_67396626809176
// MI455X (gfx1250) — compile-verified
//
#include <hip/hip_runtime.h>
#include <math.h>

// Problem constants (match reference)
#define BB 8
#define NN 4096
#define CC 64
#define KK 30
#define TINYF 1e-20f

typedef float v2f __attribute__((ext_vector_type(2)));
typedef float v8f __attribute__((ext_vector_type(8)));

// ---------------- workspace layout (bytes) ----------------
#define OFF_IDX  0                         // int  [B*N*K]   = 3,932,160 B
#define OFF_XT   (3932160)                 // float[B*N*C]   = 8,388,608 B
#define OFF_ATT  (OFF_XT + 8388608)        // float[K*B*N]   = 3,932,160 B
#define OFF_P1   (OFF_ATT + 3932160)       // float[128*32]
#define OFF_P2   (OFF_P1 + 16384)
#define OFF_P3   (OFF_P2 + 16384)
#define OFF_ST   (OFF_P3 + 16384)          // float[32] : scale/shift per BN layer
// total ~16.3 MB

// =====================================================================
// KNN: one workgroup = 16 query rows of one batch. Distances via
// V_WMMA_F32_16X16X4_F32 (A=[x,y,z,1], B=[2x,2y,2z,-|p|^2]), chunked
// 512 columns at a time into LDS; wave-per-row incremental top-30.
// =====================================================================
#define CHUNK   512
#define NCHUNK  (NN / CHUNK)
#define DSTRIDE 516   // pad so rows m and m+8 land on disjoint banks

__global__ __launch_bounds__(512)
void knn_kernel(const float* __restrict__ xloc, int* __restrict__ idxOut)
{
    __shared__ float dist[16 * DSTRIDE];      // 33,024 B
    __shared__ float cval[16][2][32];         //  4,096 B
    __shared__ int   cidx[16][2][32];         //  4,096 B

    const int tid  = threadIdx.x;
    const int wv   = tid >> 5;                // wave id == row id (16 waves)
    const int lane = tid & 31;
    const int b    = blockIdx.x >> 8;         // 256 row-tiles per batch
    const int m0   = (blockIdx.x & 255) << 4; // query row base
    const float* __restrict__ pb = xloc + b * 3 * NN;

    const int l16 = lane & 15;
    const int hi  = lane >> 4;

    // A operand: rows m0..m0+15, K = (x, y, z, 1)
    const int mr = m0 + l16;
    const float ax = pb[mr], ay = pb[NN + mr], az = pb[2 * NN + mr];
    v2f A;
    A.x = hi ? az : ax;       // K=0 | K=2
    A.y = hi ? 1.0f : ay;     // K=1 | K=3

    cval[wv][0][lane] = -__builtin_inff();
    cidx[wv][0][lane] = -1;

    float* drow = dist + wv * DSTRIDE;
    int par = 0;

    for (int ch = 0; ch < NCHUNK; ++ch) {
        const int colBase = ch * CHUNK;
        __syncthreads();  // previous selection done reading dist

        // 32 column tiles per chunk; 2 tiles per wave
        #pragma unroll
        for (int tt = 0; tt < 2; ++tt) {
            const int tile = wv * 2 + tt;
            const int n = colBase + tile * 16 + l16;
            const float bx = pb[n], by = pb[NN + n], bz = pb[2 * NN + n];
            const float xx = bx * bx + by * by + bz * bz;
            v2f Bv;
            Bv.x = hi ? 2.0f * bz : 2.0f * bx;   // K=0 | K=2
            Bv.y = hi ? -xx : 2.0f * by;         // K=1 | K=3
            v8f Cv = {};
            Cv = __builtin_amdgcn_wmma_f32_16x16x4_f32(
                    false, A, false, Bv, (short)0, Cv, false, false);
            const int cw = tile * 16 + l16;
            const int rb = hi ? 8 : 0;
            #pragma unroll
            for (int r = 0; r < 8; ++r)
                dist[(rb + r) * DSTRIDE + cw] = Cv[r];
        }
        __syncthreads();

        // wave `wv` selects top-30 of (running candidates U this chunk's row)
        float ov = cval[wv][par][lane];
        int   oi = cidx[wv][par][lane];
        float bv; int bi; int bslot;
        auto localBest = [&]() {
            bv = ov; bi = oi; bslot = -1;
            #pragma unroll
            for (int t = 0; t < CHUNK / 32; ++t) {
                const int j = lane + (t << 5);
                const float v = drow[j];
                const int gi = colBase + j;
                if (v > bv || (v == bv && (unsigned)gi < (unsigned)bi)) {
                    bv = v; bi = gi; bslot = j;
                }
            }
        };
        localBest();
        const int np = par ^ 1;
        for (int t = 0; t < KK; ++t) {
            float wvv = bv; unsigned wix = (unsigned)bi;
            #pragma unroll
            for (int off = 16; off > 0; off >>= 1) {
                const float    oV = __shfl_xor(wvv, off, 32);
                const unsigned oI = (unsigned)__shfl_xor((int)wix, off, 32);
                if (oV > wvv || (oV == wvv && oI < wix)) { wvv = oV; wix = oI; }
            }
            if (lane == 0) { cval[wv][np][t] = wvv; cidx[wv][np][t] = (int)wix; }
            if ((unsigned)bi == wix) {          // this lane held the winner
                if (bslot >= 0) drow[bslot] = -__builtin_inff();
                else { ov = -__builtin_inff(); oi = -1; }
                localBest();
            }
        }
        if (lane >= KK) { cval[wv][np][lane] = -__builtin_inff(); cidx[wv][np][lane] = -1; }
        par = np;
    }

    if (lane < KK)
        idxOut[(b * NN + m0 + wv) * KK + lane] = cidx[wv][par][lane];
}

// =====================================================================
// x [B,C,N] -> xT [B,N,C]  (LDS-tiled 32x32 transpose)
// =====================================================================
__global__ __launch_bounds__(256)
void transpose_kernel(const float* __restrict__ x, float* __restrict__ xT)
{
    __shared__ float tile[32][33];
    const int b  = blockIdx.z;
    const int c0 = blockIdx.y * 32;
    const int n0 = blockIdx.x * 32;
    const int tx = threadIdx.x, ty = threadIdx.y;  // 32 x 8
    #pragma unroll
    for (int p = 0; p < 4; ++p)
        tile[ty + 8 * p][tx] = x[(b * CC + c0 + ty + 8 * p) * NN + n0 + tx];
    __syncthreads();
    #pragma unroll
    for (int p = 0; p < 4; ++p)
        xT[(b * NN + n0 + ty + 8 * p) * CC + c0 + tx] = tile[tx][ty + 8 * p];
}

// =====================================================================
// shared device helpers
// =====================================================================
__device__ __forceinline__ void compute_feat(const float* __restrict__ p,
                                             float cx, float cy, float cz,
                                             int m, float f[6])
{
    const float xr = p[m]          - cx;
    const float yr = p[NN + m]     - cy;
    const float zr = p[2 * NN + m] - cz;
    const float sxy2 = xr * xr + yr * yr;
    const float r2   = sxy2 + zr * zr;
    const float rho = sqrtf(fmaxf(r2, TINYF));
    const float sxy = sqrtf(fmaxf(sxy2, TINYF));
    const bool dr = r2 < TINYF;
    const bool dp = sxy2 < TINYF;
    const float theta = atan2f(dr ? 0.f : zr, dr ? 1.f : sxy);
    const float phi   = atan2f(dp ? 0.f : yr, dp ? 1.f : xr);
    const float mu = (rho + theta + phi) * (1.f / 3.f);
    f[0] = rho; f[1] = theta; f[2] = phi;
    f[3] = rho - mu; f[4] = theta - mu; f[5] = phi - mu;
}

__device__ __forceinline__ void mlp_layer6(const float f[6], const float* __restrict__ W,
                                           const float* __restrict__ scale,
                                           const float* __restrict__ shift, float o[6])
{
    #pragma unroll
    for (int c = 0; c < 6; ++c) {
        float h = 0.f;
        #pragma unroll
        for (int j = 0; j < 6; ++j) h = fmaf(f[j], W[c * 6 + j], h);
        h = h * scale[c] + shift[c];
        o[c] = h > 0.f ? h : 0.2f * h;
    }
}

__device__ __forceinline__ void mlp_layer3(const float a[6], const float* __restrict__ W,
                                           const float* __restrict__ scale,
                                           const float* __restrict__ shift, float o[3])
{
    #pragma unroll
    for (int c = 0; c < 3; ++c) {
        float h = 0.f;
        #pragma unroll
        for (int j = 0; j < 6; ++j) h = fmaf(a[j], W[c * 6 + j], h);
        h = h * scale[c] + shift[c];
        o[c] = h > 0.f ? h : 0.2f * h;
    }
}

// =====================================================================
// Moment passes: accumulate per-channel sums + second moments of the
// activation entering each BN layer. BN stats derived analytically:
// mean(h)=W*mu(f), E[h^2]=w^T E[ffT] w  (h = f @ W^T is linear).
// MODE 1: feat (6)   MODE 2: a1 (6)   MODE 3: a2 (3)
// =====================================================================
template <int MODE>
__global__ __launch_bounds__(256)
void stat_pass_kernel(const float* __restrict__ xloc, const int* __restrict__ idx,
                      const float* __restrict__ W1, const float* __restrict__ W2,
                      const float* __restrict__ stats, float* __restrict__ part)
{
    __shared__ float pls[3 * NN];   // 48 KB: whole batch point cloud
    __shared__ float rbuf[256];
    const int tid = threadIdx.x;
    const int b = blockIdx.x >> 4;
    const int n = ((blockIdx.x & 15) << 8) + tid;
    const float* __restrict__ src = xloc + b * 3 * NN;
    for (int i = tid; i < 3 * NN; i += 256) pls[i] = src[i];
    __syncthreads();

    constexpr int DIM = (MODE == 3) ? 3 : 6;
    constexpr int NM  = DIM + DIM * (DIM + 1) / 2;
    float s[NM];
    #pragma unroll
    for (int i = 0; i < NM; ++i) s[i] = 0.f;

    const float cx = pls[n], cy = pls[NN + n], cz = pls[2 * NN + n];
    const int* __restrict__ ip = idx + (b * NN + n) * KK;
    for (int k = 0; k < KK; ++k) {
        float f[6];
        compute_feat(pls, cx, cy, cz, ip[k], f);
        float v[6];
        if (MODE == 1) {
            #pragma unroll
            for (int j = 0; j < 6; ++j) v[j] = f[j];
        } else if (MODE == 2) {
            mlp_layer6(f, W1, stats, stats + 6, v);
        } else {
            float a1[6];
            mlp_layer6(f, W1, stats, stats + 6, a1);
            mlp_layer3(a1, W2, stats + 12, stats + 15, v);
        }
        int q = DIM;
        #pragma unroll
        for (int i = 0; i < DIM; ++i) {
            s[i] += v[i];
            #pragma unroll
            for (int j = i; j < DIM; ++j) { s[q] += v[i] * v[j]; ++q; }
        }
    }

    // deterministic fixed-order block reduction
    #pragma unroll
    for (int q = 0; q < NM; ++q) {
        rbuf[tid] = s[q]; __syncthreads();
        for (int st = 128; st > 0; st >>= 1) {
            if (tid < st) rbuf[tid] += rbuf[tid + st];
            __syncthreads();
        }
        if (tid == 0) part[blockIdx.x * 32 + q] = rbuf[0];
        __syncthreads();
    }
}

// =====================================================================
// Finalize BN layer: collapse batch-stat normalization + affine into
// per-channel scale/shift.   statsOut: [scale[cout], shift[cout]]
// =====================================================================
__global__ void finalize_kernel(const float* __restrict__ part, int nblk, int cin, int cout,
                                const float* __restrict__ W, const float* __restrict__ g,
                                const float* __restrict__ bch, float* __restrict__ statsOut)
{
    __shared__ float sm[32];
    const int t = threadIdx.x;
    const int nm = cin + cin * (cin + 1) / 2;
    if (t < nm) {
        float a = 0.f;
        for (int i = 0; i < nblk; ++i) a += part[i * 32 + t];
        sm[t] = a;
    }
    __syncthreads();
    if (t == 0) {
        const float invN = 1.0f / (float)(BB * NN * KK);
        float mu[6];
        float E[6][6];
        for (int j = 0; j < cin; ++j) mu[j] = sm[j] * invN;
        int q = cin;
        for (int i = 0; i < cin; ++i)
            for (int j = i; j < cin; ++j) {
                const float e = sm[q] * invN; ++q;
                E[i][j] = e; E[j][i] = e;
            }
        for (int c = 0; c < cout; ++c) {
            float m = 0.f;
            for (int j = 0; j < cin; ++j) m += W[c * cin + j] * mu[j];
            float e2 = 0.f;
            for (int i = 0; i < cin; ++i)
                for (int j = 0; j < cin; ++j)
                    e2 += W[c * cin + i] * W[c * cin + j] * E[i][j];
            const float var = e2 - m * m;
            const float sc = g[c] * rsqrtf(var + 1e-5f);
            statsOut[c] = sc;
            statsOut[cout + c] = bch[c] - m * sc;
        }
    }
}

// =====================================================================
// Attention logits + softmax over k.  att layout: [k][B*N] (coalesced)
// =====================================================================
__global__ __launch_bounds__(128)
void att_kernel(const float* __restrict__ xloc, const int* __restrict__ idx,
                const float* __restrict__ W1, const float* __restrict__ W2,
                const float* __restrict__ W3, const float* __restrict__ stats,
                float* __restrict__ att)
{
    __shared__ float lbuf[KK * 128];   // 15,360 B
    const int tid = threadIdx.x;
    const int b = blockIdx.x >> 5;
    const int n = ((blockIdx.x & 31) << 7) + tid;
    const int bn = b * NN + n;
    const float* __restrict__ pb = xloc + b * 3 * NN;
    const float cx = pb[n], cy = pb[NN + n], cz = pb[2 * NN + n];
    const int* __restrict__ ip = idx + bn * KK;

    for (int k = 0; k < KK; ++k) {
        float f[6];
        compute_feat(pb, cx, cy, cz, ip[k], f);
        float a1[6];
        mlp_layer6(f, W1, stats, stats + 6, a1);
        float a2[3];
        mlp_layer3(a1, W2, stats + 12, stats + 15, a2);
        float h = a2[0] * W3[0] + a2[1] * W3[1] + a2[2] * W3[2];
        h = h * stats[18] + stats[19];
        lbuf[k * 128 + tid] = h > 0.f ? h : 0.2f * h;
    }
    float mx = -__builtin_inff();
    for (int k = 0; k < KK; ++k) mx = fmaxf(mx, lbuf[k * 128 + tid]);
    float ssum = 0.f;
    for (int k = 0; k < KK; ++k) {
        const float e = __expf(lbuf[k * 128 + tid] - mx);
        lbuf[k * 128 + tid] = e;
        ssum += e;
    }
    const float inv = 1.f / ssum;
    for (int k = 0; k < KK; ++k)
        att[k * (BB * NN) + bn] = lbuf[k * 128 + tid] * inv;
}

// =====================================================================
// Weighted neighbor aggregation (contiguous 256B gathers from xT)
// out[b,c,n] = x[b,c,n] + sum_k att * xT[b, idx, c]
// =====================================================================
__global__ __launch_bounds__(128)
void agg_kernel(const float* __restrict__ x, const int* __restrict__ idx,
                const float* __restrict__ xT, const float* __restrict__ att,
                float* __restrict__ out)
{
    const int tid = threadIdx.x;
    const int b = blockIdx.x >> 5;
    const int n = ((blockIdx.x & 31) << 7) + tid;
    const int bn = b * NN + n;
    float4 acc[16];
    #pragma unroll
    for (int i = 0; i < 16; ++i) acc[i] = make_float4(0.f, 0.f, 0.f, 0.f);
    const int* __restrict__ ip = idx + bn * KK;
    for (int k = 0; k < KK; ++k) {
        const float w = att[k * (BB * NN) + bn];
        const float4* __restrict__ xp = (const float4*)(xT + (size_t)(b * NN + ip[k]) * CC);
        #pragma unroll
        for (int i = 0; i < 16; ++i) {
            const float4 v = xp[i];
            acc[i].x = fmaf(w, v.x, acc[i].x);
            acc[i].y = fmaf(w, v.y, acc[i].y);
            acc[i].z = fmaf(w, v.z, acc[i].z);
            acc[i].w = fmaf(w, v.w, acc[i].w);
        }
    }
    const float* __restrict__ xin = x + (size_t)b * CC * NN + n;
    float* __restrict__ op = out + (size_t)b * CC * NN + n;
    #pragma unroll
    for (int i = 0; i < 16; ++i) {
        op[(4 * i + 0) * NN] = xin[(4 * i + 0) * NN] + acc[i].x;
        op[(4 * i + 1) * NN] = xin[(4 * i + 1) * NN] + acc[i].y;
        op[(4 * i + 2) * NN] = xin[(4 * i + 2) * NN] + acc[i].z;
        op[(4 * i + 3) * NN] = xin[(4 * i + 3) * NN] + acc[i].w;
    }
}

// =====================================================================
extern "C" void kernel_launch(void* const* d_in, const int* in_sizes, int n_in,
                              void* d_out, int out_size, void* d_ws, size_t ws_size,
                              hipStream_t stream)
{
    (void)in_sizes; (void)n_in; (void)out_size; (void)ws_size;
    const float* xloc = (const float*)d_in[0];
    const float* x    = (const float*)d_in[1];
    const float* W1   = (const float*)d_in[2];
    const float* g1   = (const float*)d_in[3];
    const float* b1   = (const float*)d_in[4];
    const float* W2   = (const float*)d_in[5];
    const float* g2   = (const float*)d_in[6];
    const float* b2   = (const float*)d_in[7];
    const float* W3   = (const float*)d_in[8];
    const float* g3   = (const float*)d_in[9];
    const float* b3   = (const float*)d_in[10];
    float* out = (float*)d_out;
    char*  ws  = (char*)d_ws;

    int*   idx   = (int*)  (ws + OFF_IDX);
    float* xT    = (float*)(ws + OFF_XT);
    float* att   = (float*)(ws + OFF_ATT);
    float* p1    = (float*)(ws + OFF_P1);
    float* p2    = (float*)(ws + OFF_P2);
    float* p3    = (float*)(ws + OFF_P3);
    float* stats = (float*)(ws + OFF_ST);

    knn_kernel<<<dim3(BB * (NN / 16)), dim3(512), 0, stream>>>(xloc, idx);
    transpose_kernel<<<dim3(NN / 32, CC / 32, BB), dim3(32, 8), 0, stream>>>(x, xT);

    stat_pass_kernel<1><<<dim3(128), dim3(256), 0, stream>>>(xloc, idx, W1, W2, stats, p1);
    finalize_kernel<<<dim3(1), dim3(32), 0, stream>>>(p1, 128, 6, 6, W1, g1, b1, stats);
    stat_pass_kernel<2><<<dim3(128), dim3(256), 0, stream>>>(xloc, idx, W1, W2, stats, p2);
    finalize_kernel<<<dim3(1), dim3(32), 0, stream>>>(p2, 128, 6, 3, W2, g2, b2, stats + 12);
    stat_pass_kernel<3><<<dim3(128), dim3(256), 0, stream>>>(xloc, idx, W1, W2, stats, p3);
    finalize_kernel<<<dim3(1), dim3(32), 0, stream>>>(p3, 128, 3, 1, W3, g3, b3, stats + 18);

    att_kernel<<<dim3(BB * 32), dim3(128), 0, stream>>>(xloc, idx, W1, W2, W3, stats, att);
    agg_kernel<<<dim3(BB * 32), dim3(128), 0, stream>>>(x, idx, xT, att, out);
}